// MPS_84370337562944
// MI455X (gfx1250) — compile-verified
//
#include <hip/hip_runtime.h>
#include <hip/hip_bf16.h>

// ---------------------------------------------------------------------------
// MPS log-likelihood on MI455X (gfx1250, wave32)
//   inputs: x (1024,512) int64 ; g0 (1,128,32) f32 ; g_mid (510,32,128,32) f32
//           g_last (32,128,1) f32
//   output: scalar f32 loss
// ---------------------------------------------------------------------------

#define N_CORES 512
#define N_MID   510
#define PHYS_D  128
#define BOND_R  32
#define BATCH   1024

#define NORM_THREADS 512            // 16 waves, 4 per SIMD32
#define NORM_WAVES   (NORM_THREADS / 32)
#define D_PER_WAVE   (PHYS_D / NORM_WAVES)
// dynamic LDS: sL[1024] | sC[1024] | sT[NORM_WAVES][1024] | sRed[NORM_WAVES]
#define NORM_SMEM_FLOATS (1024 + 1024 + NORM_WAVES * 1024 + NORM_WAVES)

typedef __attribute__((ext_vector_type(2))) float v2f;
typedef __attribute__((ext_vector_type(8))) float v8f;

// --- WMMA helpers: 16x16x16 tile matmul built from 4 chained f32 16x16x4 ----
// A-fragment (32-bit A 16x4): lanes 0-15 -> M=lane, v0=K(+0), v1=K(+1);
//                             lanes 16-31 -> v0=K(+2), v1=K(+3).
// B-fragment (4x16): v0 = rows {k, k+2} striped across lane halves, v1 = {k+1, k+3}.
// C/D: VGPR r -> row r (lanes 0-15) / row r+8 (lanes 16-31).

__device__ __forceinline__ v8f mm16_AB(const float* __restrict__ A, int lda,
                                       const float* __restrict__ B, int ldb,
                                       v8f acc, int l16, int half) {
#pragma unroll
  for (int k0 = 0; k0 < 16; k0 += 4) {
    const int k = k0 + 2 * half;
    v2f a, b;
    a.x = A[l16 * lda + k];
    a.y = A[l16 * lda + k + 1];
    b.x = B[k * ldb + l16];
    b.y = B[(k + 1) * ldb + l16];
    acc = __builtin_amdgcn_wmma_f32_16x16x4_f32(false, a, false, b,
                                                (short)0, acc, false, false);
  }
  return acc;
}

// acc += A^T * B  (A stored 16x16 row-major with stride lda; read transposed)
__device__ __forceinline__ v8f mm16_ATB(const float* __restrict__ A, int lda,
                                        const float* __restrict__ B, int ldb,
                                        v8f acc, int l16, int half) {
#pragma unroll
  for (int k0 = 0; k0 < 16; k0 += 4) {
    const int k = k0 + 2 * half;
    v2f a, b;
    a.x = A[k * lda + l16];        // A^T[l16][k]
    a.y = A[(k + 1) * lda + l16];
    b.x = B[k * ldb + l16];
    b.y = B[(k + 1) * ldb + l16];
    acc = __builtin_amdgcn_wmma_f32_16x16x4_f32(false, a, false, b,
                                                (short)0, acc, false, false);
  }
  return acc;
}

__device__ __forceinline__ void store_tile(float* T, int ldt, v8f d,
                                           int l16, int half) {
#pragma unroll
  for (int r = 0; r < 8; ++r)
    T[(r + 8 * half) * ldt + l16] = d[r];
}

// ---------------------------------------------------------------------------
// Kernel 0: zero the workspace accumulators (ws[0]=sum log|psi|, ws[1]=count,
//           ws[2]=log z + sum log sf)
// ---------------------------------------------------------------------------
__global__ void mps_init_ws(float* ws) {
  if (threadIdx.x < 4) ws[threadIdx.x] = 0.0f;
}

// ---------------------------------------------------------------------------
// Kernel 1: select chain. One wave32 per batch element, lane j owns psi[j].
// 4 independent accumulators break the 32-term serial fma chain.
// ---------------------------------------------------------------------------
__global__ __launch_bounds__(256) void mps_select_kernel(
    const long long* __restrict__ x, const float* __restrict__ g0,
    const float* __restrict__ gmid, const float* __restrict__ glast,
    float* ws) {
  const int wave = threadIdx.x >> 5;
  const int lane = threadIdx.x & 31;
  const int b = blockIdx.x * 8 + wave;
  const long long* xb = x + (size_t)b * N_CORES;

  // psi0[j] = g0[0, x[b,0], j]
  float psi = g0[(int)xb[0] * BOND_R + lane];

  int d_next = (int)xb[1];
  for (int c = 0; c < N_MID; ++c) {
    const float* base =
        gmid + (size_t)c * (BOND_R * PHYS_D * BOND_R) + (size_t)d_next * BOND_R;
    // prefetch next core's gathered 32x32 matrix while we work on this one
    if (c + 1 < N_MID) {
      d_next = (int)xb[c + 2];
      const float* nbase = gmid + (size_t)(c + 1) * (BOND_R * PHYS_D * BOND_R) +
                           (size_t)d_next * BOND_R;
      __builtin_prefetch(nbase + (size_t)lane * (PHYS_D * BOND_R), 0, 1);
    }
    float a0 = 0.0f, a1 = 0.0f, a2 = 0.0f, a3 = 0.0f;
#pragma unroll
    for (int i = 0; i < BOND_R; i += 4) {
      a0 = fmaf(__shfl(psi, i + 0, 32),
                base[(size_t)(i + 0) * (PHYS_D * BOND_R) + lane], a0);
      a1 = fmaf(__shfl(psi, i + 1, 32),
                base[(size_t)(i + 1) * (PHYS_D * BOND_R) + lane], a1);
      a2 = fmaf(__shfl(psi, i + 2, 32),
                base[(size_t)(i + 2) * (PHYS_D * BOND_R) + lane], a2);
      a3 = fmaf(__shfl(psi, i + 3, 32),
                base[(size_t)(i + 3) * (PHYS_D * BOND_R) + lane], a3);
    }
    psi = (a0 + a1) + (a2 + a3);
  }

  // psi_b = sum_i psi[i] * g_last[i, x[b,511], 0]
  float v = psi * glast[lane * PHYS_D + (int)xb[N_CORES - 1]];
#pragma unroll
  for (int off = 16; off; off >>= 1) v += __shfl_down(v, off, 32);

  if (lane == 0) {
    const float lv = logf(fabsf(v));
    if (lv == lv) {  // nanmean: skip NaN
      atomicAdd(&ws[0], lv);
      atomicAdd(&ws[1], 1.0f);
    }
  }
}

// ---------------------------------------------------------------------------
// Kernel 2: norm chain. Single workgroup (16 waves = 4/SIMD for WMMA overlap),
// sequential over 510 cores; per core: L_new = sum_d A_d^T L A_d via WMMA
// f32 16x16x4, then Frobenius rescale.  Dynamic LDS (~74 KB; WGP has 320 KB).
// ---------------------------------------------------------------------------
__global__ __launch_bounds__(NORM_THREADS) void mps_norm_kernel(
    const float* __restrict__ g0, const float* __restrict__ gmid,
    const float* __restrict__ glast, float* ws) {
  extern __shared__ float smem[];
  float* sL = smem;                       // 32x32 current L, row-major
  float* sC = smem + 1024;                // L_new accumulator
  float* sTall = smem + 2048;             // per-wave T = L*A_d scratch
  float* sRed = smem + 2048 + NORM_WAVES * 1024;

  const int t = threadIdx.x;
  const int wave = t >> 5;
  const int lane = t & 31;
  const int l16 = lane & 15;
  const int half = lane >> 4;
  float* Tw = sTall + wave * 1024;

  // L0[q][s] = sum_d g0[d][q] * g0[d][s]
  for (int e = t; e < 1024; e += NORM_THREADS) {
    const int q = e >> 5, s = e & 31;
    float a = 0.0f;
    for (int d = 0; d < PHYS_D; ++d)
      a = fmaf(g0[d * BOND_R + q], g0[d * BOND_R + s], a);
    sL[e] = a;
    sC[e] = 0.0f;
  }
  __syncthreads();

  float logsum = 0.0f;

  for (int c = 0; c < N_MID; ++c) {
    const float* gc = gmid + (size_t)c * (BOND_R * PHYS_D * BOND_R);

    v8f acc00 = {}, acc01 = {}, acc10 = {}, acc11 = {};

    for (int td = 0; td < D_PER_WAVE; ++td) {    // 16 waves x 8 d = 128
      const int d = wave * D_PER_WAVE + td;
      const float* A = gc + (size_t)d * BOND_R;  // A[i][j] = A[i*4096 + j]

      // step 1: T = L @ A_d  (2x2 tiles, K=32)
#pragma unroll
      for (int a = 0; a < 2; ++a) {
#pragma unroll
        for (int cc = 0; cc < 2; ++cc) {
          v8f acc = {};
#pragma unroll
          for (int p = 0; p < 2; ++p)
            acc = mm16_AB(sL + (a * 16) * 32 + p * 16, 32,
                          A + (size_t)(p * 16) * 4096 + cc * 16, 4096,
                          acc, l16, half);
          store_tile(Tw + (a * 16) * 32 + cc * 16, 32, acc, l16, half);
        }
      }
      // per-wave LDS is in-order: T is visible to this wave's loads below.

      // step 2: C_{q,cc} += sum_p (A_{p,q})^T @ T_{p,cc}
#pragma unroll
      for (int p = 0; p < 2; ++p) {
        const float* Ap = A + (size_t)(p * 16) * 4096;
        const float* Tp = Tw + (p * 16) * 32;
        acc00 = mm16_ATB(Ap, 4096, Tp, 32, acc00, l16, half);
        acc01 = mm16_ATB(Ap, 4096, Tp + 16, 32, acc01, l16, half);
        acc10 = mm16_ATB(Ap + 16, 4096, Tp, 32, acc10, l16, half);
        acc11 = mm16_ATB(Ap + 16, 4096, Tp + 16, 32, acc11, l16, half);
      }
    }

    // reduce the per-wave partials into sC (ds_add_f32)
#pragma unroll
    for (int r = 0; r < 8; ++r) {
      const int row = r + 8 * half;
      atomicAdd(&sC[row * 32 + l16], acc00[r]);
      atomicAdd(&sC[row * 32 + 16 + l16], acc01[r]);
      atomicAdd(&sC[(16 + row) * 32 + l16], acc10[r]);
      atomicAdd(&sC[(16 + row) * 32 + 16 + l16], acc11[r]);
    }
    __syncthreads();

    // Frobenius norm of L_new
    float part = 0.0f;
    for (int e = t; e < 1024; e += NORM_THREADS) part = fmaf(sC[e], sC[e], part);
#pragma unroll
    for (int off = 16; off; off >>= 1) part += __shfl_down(part, off, 32);
    if (lane == 0) sRed[wave] = part;
    __syncthreads();
    if (t == 0) {
      float s = 0.0f;
      for (int w = 0; w < NORM_WAVES; ++w) s += sRed[w];
      sRed[0] = sqrtf(s);
    }
    __syncthreads();
    const float sf = sRed[0];
    logsum += logf(sf);
    const float inv = 1.0f / sf;
    for (int e = t; e < 1024; e += NORM_THREADS) {
      sL[e] = sC[e] * inv;
      sC[e] = 0.0f;
    }
    __syncthreads();
  }

  // z = sum_{p,q} L[p,q] * (sum_d g_last[p,d] * g_last[q,d])
  float zpart = 0.0f;
  for (int e = t; e < 1024; e += NORM_THREADS) {
    const int p = e >> 5, q = e & 31;
    float dot = 0.0f;
    for (int d = 0; d < PHYS_D; ++d)
      dot = fmaf(glast[p * PHYS_D + d], glast[q * PHYS_D + d], dot);
    zpart = fmaf(sL[e], dot, zpart);
  }
#pragma unroll
  for (int off = 16; off; off >>= 1) zpart += __shfl_down(zpart, off, 32);
  if (lane == 0) sRed[wave] = zpart;
  __syncthreads();
  if (t == 0) {
    float z = 0.0f;
    for (int w = 0; w < NORM_WAVES; ++w) z += sRed[w];
    ws[2] = logf(z) + logsum;
  }
}

// ---------------------------------------------------------------------------
// Kernel 3: loss = (log z + sum log sf) - 2 * nanmean(log|psi|)
// ---------------------------------------------------------------------------
__global__ void mps_finalize(const float* ws, float* out) {
  out[0] = ws[2] - 2.0f * (ws[0] / fmaxf(ws[1], 1.0f));
}

// ---------------------------------------------------------------------------
extern "C" void kernel_launch(void* const* d_in, const int* in_sizes, int n_in,
                              void* d_out, int out_size, void* d_ws,
                              size_t ws_size, hipStream_t stream) {
  const long long* x = (const long long*)d_in[0];   // (1024,512) int64
  const float* g0 = (const float*)d_in[1];          // (1,128,32)
  const float* gmid = (const float*)d_in[2];        // (510,32,128,32)
  const float* glast = (const float*)d_in[3];       // (32,128,1)
  float* out = (float*)d_out;
  float* ws = (float*)d_ws;

  const int norm_smem = NORM_SMEM_FLOATS * (int)sizeof(float);  // ~74 KB
  // Opt in to >64KB dynamic LDS (WGP has 320 KB). Host-side attribute set,
  // not a stream operation: safe under graph capture, deterministic.
  (void)hipFuncSetAttribute((const void*)mps_norm_kernel,
                            hipFuncAttributeMaxDynamicSharedMemorySize,
                            norm_smem);

  mps_init_ws<<<1, 64, 0, stream>>>(ws);
  mps_select_kernel<<<BATCH / 8, 256, 0, stream>>>(x, g0, gmid, glast, ws);
  mps_norm_kernel<<<1, NORM_THREADS, norm_smem, stream>>>(g0, gmid, glast, ws);
  mps_finalize<<<1, 1, 0, stream>>>(ws, out);
}